// GroupShuffleLayer_65352222376295
// MI455X (gfx1250) — compile-verified
//
#include <hip/hip_runtime.h>
#include <hip/hip_bf16.h>
#include <stdint.h>

typedef __attribute__((ext_vector_type(2))) float v2f;
typedef __attribute__((ext_vector_type(8))) float v8f;

#define DIM 1024
#define BLK 64
#define NBLK 16          // DIM / BLK
#define EXPM_TERMS 12    // ||A||~0.16 -> term 13 ~ 0.16^13/13! << fp32 eps

// ---------------------------------------------------------------------------
// Kernel 1: per-block expm via Taylor series, output in WMMA B-fragment layout.
// Bout[k][chunk][idx]: chunk = ks*4+nt (64 chunks), idx = lane*2+v (64 floats).
// Fragment element (lane l, vgpr v) = B[ks*4 + (l>>4)*2 + v][nt*16 + (l&15)].
// ---------------------------------------------------------------------------
__global__ void __launch_bounds__(256) expm_blocks_kernel(
    const float* __restrict__ skew, float* __restrict__ Bout) {
  __shared__ float A [BLK * BLK];
  __shared__ float T0[BLK * BLK];
  __shared__ float T1[BLK * BLK];
  __shared__ float S [BLK * BLK];

  const int k = blockIdx.x;
  const int t = threadIdx.x;
  const float* P = skew + k * BLK * BLK;

  // A = P - P^T ; S = I + A ; T = A
  for (int e = t; e < BLK * BLK; e += 256) {
    const int r = e >> 6, c = e & 63;
    const float a = P[r * BLK + c] - P[c * BLK + r];
    A[e] = a;
    T0[e] = a;
    S[e] = a + ((r == c) ? 1.0f : 0.0f);
  }
  __syncthreads();

  float* Tcur = T0;
  float* Tnext = T1;
  for (int term = 2; term <= EXPM_TERMS; ++term) {
    const float inv = 1.0f / (float)term;
    for (int e = t; e < BLK * BLK; e += 256) {
      const int r = e >> 6, c = e & 63;
      float acc = 0.0f;
      #pragma unroll 8
      for (int m = 0; m < BLK; ++m)
        acc += Tcur[r * BLK + m] * A[m * BLK + c];
      acc *= inv;           // T_k = T_{k-1} @ A / k  == A^k / k!
      Tnext[e] = acc;
      S[e] += acc;          // private to this thread's e
    }
    __syncthreads();
    float* tmp = Tcur; Tcur = Tnext; Tnext = tmp;
  }

  // Emit B pre-swizzled for single-b64-per-lane fragment loads.
  for (int e = t; e < BLK * BLK; e += 256) {
    const int chunk = e >> 6;           // (ks, nt)
    const int idx   = e & 63;           // (lane, v)
    const int l = idx >> 1, v = idx & 1;
    const int ks = chunk >> 2, nt = chunk & 3;
    const int row = ks * 4 + ((l >> 4) << 1) + v;   // K index within block
    const int col = nt * 16 + (l & 15);             // N index within block
    Bout[k * BLK * BLK + e] = S[row * BLK + col];
  }
}

// ---------------------------------------------------------------------------
// Kernel 2: inverse permutation (perm is int64 from JAX).
// ---------------------------------------------------------------------------
__global__ void invperm_kernel(const long long* __restrict__ perm,
                               int* __restrict__ inv) {
  const int j = blockIdx.x * blockDim.x + threadIdx.x;
  if (j < DIM) inv[(int)perm[j]] = j;
}

// ---------------------------------------------------------------------------
// Kernel 3: block-diagonal GEMM + permuted scatter store.
// One wave per (16-row x 64-col) tile. wave32. K=64 in 16 steps of
// v_wmma_f32_16x16x4_f32 into four 16x16 f32 accumulators.
// ---------------------------------------------------------------------------
__global__ void __launch_bounds__(128) bdiag_gemm_kernel(
    const float* __restrict__ x, const float* __restrict__ Bf,
    const int* __restrict__ invp, float* __restrict__ out) {
  const int wave = blockIdx.x * (blockDim.x >> 5) + (threadIdx.x >> 5);
  const int lane = threadIdx.x & 31;

  const int k  = wave & (NBLK - 1);   // which diagonal block
  const int rt = wave >> 4;           // row tile
  const int m0 = rt << 4;             // first of 16 rows

  const int lrow = lane & 15;         // M index this lane serves in A frags, N in C frags
  const int lhi  = lane >> 4;         // lane half

  // A fragment (16x4 fp32): VGPR v, lane l -> x[m0 + (l&15)][kb + (l>>4)*2 + v]
  // -> the two per-lane values are column-adjacent: one float2 load.
  const float* xa = x + (size_t)(m0 + lrow) * DIM + k * BLK + lhi * 2;
  // B fragments: contiguous 64-float chunks, lane l reads float2 at +2*l.
  const float* bf = Bf + k * BLK * BLK + lane * 2;

  v8f acc[4] = {v8f{}, v8f{}, v8f{}, v8f{}};

  #pragma unroll 4
  for (int ks = 0; ks < 16; ++ks) {
    const v2f a = *(const v2f*)(xa + ks * 4);
    const float* bb = bf + (ks * 4) * 64;          // 4 nt-chunks of 64 floats
    const v2f b0 = *(const v2f*)(bb + 0 * 64);
    const v2f b1 = *(const v2f*)(bb + 1 * 64);
    const v2f b2 = *(const v2f*)(bb + 2 * 64);
    const v2f b3 = *(const v2f*)(bb + 3 * 64);
    acc[0] = __builtin_amdgcn_wmma_f32_16x16x4_f32(false, a, false, b0,
                                                   (short)0, acc[0], false, false);
    acc[1] = __builtin_amdgcn_wmma_f32_16x16x4_f32(false, a, false, b1,
                                                   (short)0, acc[1], false, false);
    acc[2] = __builtin_amdgcn_wmma_f32_16x16x4_f32(false, a, false, b2,
                                                   (short)0, acc[2], false, false);
    acc[3] = __builtin_amdgcn_wmma_f32_16x16x4_f32(false, a, false, b3,
                                                   (short)0, acc[3], false, false);
  }

  // C/D layout: VGPR r, lanes 0-15 -> M=r, lanes 16-31 -> M=r+8; N = lane&15.
  // Scatter through inverse permutation: out[:, invp[c]] = y[:, c].
  const int cbase = k * BLK + lrow;
  float* orow = out + (size_t)(m0 + 8 * lhi) * DIM;
  #pragma unroll
  for (int nt = 0; nt < 4; ++nt) {
    const int j = invp[cbase + nt * 16];
    float* o = orow + j;
    #pragma unroll
    for (int r = 0; r < 8; ++r)
      o[(size_t)r * DIM] = acc[nt][r];
  }
}

// ---------------------------------------------------------------------------
extern "C" void kernel_launch(void* const* d_in, const int* in_sizes, int n_in,
                              void* d_out, int out_size, void* d_ws, size_t ws_size,
                              hipStream_t stream) {
  const float* x = (const float*)d_in[0];          // [N, 1024] fp32
  const float* skew = (const float*)d_in[1];       // [16, 64, 64] fp32
  const long long* perm = (const long long*)d_in[2]; // [1024] int64
  float* out = (float*)d_out;

  float* Bf = (float*)d_ws;                                  // 16*64*64 floats
  int* invp = (int*)((char*)d_ws + NBLK * BLK * BLK * sizeof(float));

  const int rows = in_sizes[0] / DIM;              // 65536
  const int row_tiles = rows / 16;                 // 4096
  const int total_waves = row_tiles * NBLK;        // 65536
  const int waves_per_block = 4;                   // 128 threads, wave32

  expm_blocks_kernel<<<NBLK, 256, 0, stream>>>(skew, Bf);
  invperm_kernel<<<(DIM + 255) / 256, 256, 0, stream>>>(perm, invp);
  bdiag_gemm_kernel<<<total_waves / waves_per_block, waves_per_block * 32, 0,
                      stream>>>(x, Bf, invp, out);
}